// GAE_17755394801760
// MI455X (gfx1250) — compile-verified
//
#include <hip/hip_runtime.h>
#include <hip/hip_bf16.h>

typedef __attribute__((ext_vector_type(2))) float v2f;
typedef __attribute__((ext_vector_type(8))) float v8f;

#define DD 512
#define HH 8
#define LN_EPS 1e-5f

// ---------------------------------------------------------------------------
// init: deg = 1.0 (self-loop), agg1 = agg2 = 0
// ---------------------------------------------------------------------------
__global__ void init_kernel(float* __restrict__ deg, float* __restrict__ agg1,
                            float* __restrict__ agg2, int N) {
    int i = blockIdx.x * blockDim.x + threadIdx.x;
    if (i < N) deg[i] = 1.0f;
    if (i < 8 * N) { agg1[i] = 0.0f; agg2[i] = 0.0f; }
}

// ---------------------------------------------------------------------------
// degree accumulation over edges (dst side)
// ---------------------------------------------------------------------------
__global__ void deg_kernel(const long long* __restrict__ ei, float* __restrict__ deg,
                           int E) {
    int e = blockIdx.x * blockDim.x + threadIdx.x;
    if (e >= E) return;
    atomicAdd(&deg[(int)ei[(long long)E + e]], 1.0f);
}

// deg -> rsqrt(deg) in place
__global__ void dinv_kernel(float* __restrict__ deg, int N) {
    int n = blockIdx.x * blockDim.x + threadIdx.x;
    if (n < N) deg[n] = rsqrtf(deg[n]);
}

// ---------------------------------------------------------------------------
// GEMM1: h1 = x @ W1   (N x 512) @ (512 x 8) via V_WMMA_F32_16X16X4_F32.
// W1 staged zero-padded (512 x 16) in LDS so B fragments are unconditional
// DS loads (no EXEC-masked branches in the hot loop).
// A frag (16x4 f32): lane holds {A[lane%16][koff], A[lane%16][koff+1]},
//   koff = (lane<16)?0:2.  B frag (4x16): lane holds {B[koff][lane%16], ...}.
// C/D: VGPR v = rows v / v+8 (lane<16 / lane>=16), col = lane%16.
// ---------------------------------------------------------------------------
__global__ __launch_bounds__(256) void gemm1_wmma_kernel(
    const float* __restrict__ x, const float* __restrict__ W1,
    float* __restrict__ h1, int N) {
    __shared__ float W1s[DD * 16];               // zero-padded: cols 8..15 = 0

    for (int i = threadIdx.x; i < DD * 16; i += 256) {
        int kk = i >> 4;
        int cc = i & 15;
        W1s[i] = (cc < HH) ? W1[kk * HH + cc] : 0.0f;
    }
    __syncthreads();

    const int lane = threadIdx.x & 31;
    const int wave = threadIdx.x >> 5;
    const int tile = blockIdx.x * 8 + wave;
    const int row0 = tile * 16;
    if (row0 >= N) return;

    const int m    = lane & 15;
    const int koff = (lane < 16) ? 0 : 2;
    const int col  = lane & 15;

    int arow = row0 + m;
    if (arow > N - 1) arow = N - 1;              // tail clamp (N%16==0 normally)
    const float* __restrict__ xrow = x + (size_t)arow * DD;
    const float* __restrict__ brow = W1s + (size_t)koff * 16 + col;

    v8f c = {};
#pragma unroll 4
    for (int k = 0; k < DD; k += 4) {
        v2f a = *(const v2f*)(xrow + k + koff);  // global_load_b64
        v2f b;
        b.x = brow[k * 16];                      // ds_load, conflict-free
        b.y = brow[k * 16 + 16];
        c = __builtin_amdgcn_wmma_f32_16x16x4_f32(false, a, false, b,
                                                  (short)0, c, false, false);
    }

    if (col < HH) {
        const int rbase = row0 + ((lane < 16) ? 0 : 8);
        if (row0 + 16 <= N) {                    // uniform fast path
#pragma unroll
            for (int v = 0; v < 8; ++v)
                h1[(size_t)(rbase + v) * HH + col] = c[v];
        } else {
#pragma unroll
            for (int v = 0; v < 8; ++v)
                if (rbase + v < N) h1[(size_t)(rbase + v) * HH + col] = c[v];
        }
    }
}

// ---------------------------------------------------------------------------
// scatter-add in the H=8 bottleneck dim: acc[dst] += feat[src] * dinv[s]*dinv[d]
// 8 threads per edge; accumulators are L2-resident (1.6 MB).
// ---------------------------------------------------------------------------
__global__ void scatter_kernel(const long long* __restrict__ ei,
                               const float* __restrict__ dinv,
                               const float* __restrict__ feat,
                               float* __restrict__ acc, int E) {
    int t = blockIdx.x * blockDim.x + threadIdx.x;
    if (t >= E * 8) return;
    int e = t >> 3;
    int j = t & 7;
    int s = (int)ei[e];
    int d = (int)ei[(long long)E + e];
    float w = dinv[s] * dinv[d];
    atomicAdd(&acc[d * 8 + j], feat[s * 8 + j] * w);
}

// ---------------------------------------------------------------------------
// bottleneck epilogue: add self-loop + bias, LayerNorm(H=8), ReLU -> z
// ---------------------------------------------------------------------------
__global__ void ln_relu_kernel(const float* __restrict__ agg1,
                               const float* __restrict__ h1,
                               const float* __restrict__ dinv,
                               const float* __restrict__ b1,
                               const float* __restrict__ gamma,
                               const float* __restrict__ beta,
                               float* __restrict__ z, int N) {
    int n = blockIdx.x * blockDim.x + threadIdx.x;
    if (n >= N) return;
    float di = dinv[n];
    float selfw = di * di;

    float4 a0 = *(const float4*)(agg1 + (size_t)n * 8);
    float4 a1 = *(const float4*)(agg1 + (size_t)n * 8 + 4);
    float4 s0 = *(const float4*)(h1 + (size_t)n * 8);
    float4 s1 = *(const float4*)(h1 + (size_t)n * 8 + 4);

    float h[8];
    h[0] = a0.x + s0.x * selfw + b1[0];
    h[1] = a0.y + s0.y * selfw + b1[1];
    h[2] = a0.z + s0.z * selfw + b1[2];
    h[3] = a0.w + s0.w * selfw + b1[3];
    h[4] = a1.x + s1.x * selfw + b1[4];
    h[5] = a1.y + s1.y * selfw + b1[5];
    h[6] = a1.z + s1.z * selfw + b1[6];
    h[7] = a1.w + s1.w * selfw + b1[7];

    float mu = 0.0f;
#pragma unroll
    for (int j = 0; j < 8; ++j) mu += h[j];
    mu *= 0.125f;
    float var = 0.0f;
#pragma unroll
    for (int j = 0; j < 8; ++j) {
        float dlt = h[j] - mu;
        var += dlt * dlt;
    }
    var *= 0.125f;
    float rs = rsqrtf(var + LN_EPS);

    float4 o0, o1;
    o0.x = fmaxf((h[0] - mu) * rs * gamma[0] + beta[0], 0.0f);
    o0.y = fmaxf((h[1] - mu) * rs * gamma[1] + beta[1], 0.0f);
    o0.z = fmaxf((h[2] - mu) * rs * gamma[2] + beta[2], 0.0f);
    o0.w = fmaxf((h[3] - mu) * rs * gamma[3] + beta[3], 0.0f);
    o1.x = fmaxf((h[4] - mu) * rs * gamma[4] + beta[4], 0.0f);
    o1.y = fmaxf((h[5] - mu) * rs * gamma[5] + beta[5], 0.0f);
    o1.z = fmaxf((h[6] - mu) * rs * gamma[6] + beta[6], 0.0f);
    o1.w = fmaxf((h[7] - mu) * rs * gamma[7] + beta[7], 0.0f);
    *(float4*)(z + (size_t)n * 8)     = o0;
    *(float4*)(z + (size_t)n * 8 + 4) = o1;
}

// ---------------------------------------------------------------------------
// GEMM2: out = relu((agg2 + z*dinv^2) @ W2 + b2) * size_factors
// (N x 8) @ (8 x 512); one wave per 16-row tile, loops 32 column tiles,
// 2 WMMA f32 16x16x4 per column tile (K = 8). W2 staged in LDS with padded
// row stride (520) so both half-waves hit disjoint bank sets.
// ---------------------------------------------------------------------------
#define W2_STRIDE 520

__global__ __launch_bounds__(256) void gemm2_wmma_kernel(
    const float* __restrict__ agg2, const float* __restrict__ z,
    const float* __restrict__ dinv, const float* __restrict__ W2,
    const float* __restrict__ b2, const float* __restrict__ sf,
    float* __restrict__ out, int N) {
    __shared__ float W2s[HH * W2_STRIDE];

    for (int i = threadIdx.x; i < HH * DD; i += 256) {
        int kk = i >> 9;
        int cc = i & (DD - 1);
        W2s[kk * W2_STRIDE + cc] = W2[i];
    }
    __syncthreads();

    const int lane = threadIdx.x & 31;
    const int wave = threadIdx.x >> 5;
    const int tile = blockIdx.x * 8 + wave;
    const int row0 = tile * 16;
    if (row0 >= N) return;

    const int m    = lane & 15;
    const int koff = (lane < 16) ? 0 : 2;

    int r = row0 + m;
    if (r > N - 1) r = N - 1;
    float di = dinv[r];
    float selfw = di * di;

    // A fragments for K = 0..3 and K = 4..7, with self-loop folded in.
    v2f a0, a1;
    {
        float4 g0 = *(const float4*)(agg2 + (size_t)r * 8);
        float4 g1 = *(const float4*)(agg2 + (size_t)r * 8 + 4);
        float4 z0 = *(const float4*)(z + (size_t)r * 8);
        float4 z1 = *(const float4*)(z + (size_t)r * 8 + 4);
        float ag[8] = {g0.x + z0.x * selfw, g0.y + z0.y * selfw,
                       g0.z + z0.z * selfw, g0.w + z0.w * selfw,
                       g1.x + z1.x * selfw, g1.y + z1.y * selfw,
                       g1.z + z1.z * selfw, g1.w + z1.w * selfw};
        a0.x = ag[0 + koff];
        a0.y = ag[1 + koff];
        a1.x = ag[4 + koff];
        a1.y = ag[5 + koff];
    }

    const int rbase = row0 + ((lane < 16) ? 0 : 8);
    const bool fullTile = (row0 + 16 <= N);
    const float* __restrict__ bcol = W2s + (size_t)koff * W2_STRIDE + m;

#pragma unroll 2
    for (int ct = 0; ct < DD / 16; ++ct) {
        const int ncol = ct * 16 + m;
        v2f b0, b1v;
        b0.x  = bcol[ct * 16];                          // ds_load, row koff
        b0.y  = bcol[ct * 16 + W2_STRIDE];              // row koff+1
        b1v.x = bcol[ct * 16 + 4 * W2_STRIDE];          // row koff+4
        b1v.y = bcol[ct * 16 + 5 * W2_STRIDE];          // row koff+5

        v8f c = {};
        c = __builtin_amdgcn_wmma_f32_16x16x4_f32(false, a0, false, b0,
                                                  (short)0, c, false, false);
        c = __builtin_amdgcn_wmma_f32_16x16x4_f32(false, a1, false, b1v,
                                                  (short)0, c, false, false);

        float bias = b2[ncol];
        if (fullTile) {
#pragma unroll
            for (int v = 0; v < 8; ++v) {
                int rr = rbase + v;
                out[(size_t)rr * DD + ncol] = fmaxf(c[v] + bias, 0.0f) * sf[rr];
            }
        } else {
#pragma unroll
            for (int v = 0; v < 8; ++v) {
                int rr = rbase + v;
                if (rr < N)
                    out[(size_t)rr * DD + ncol] = fmaxf(c[v] + bias, 0.0f) * sf[rr];
            }
        }
    }
}

// ---------------------------------------------------------------------------
// launch
// ---------------------------------------------------------------------------
extern "C" void kernel_launch(void* const* d_in, const int* in_sizes, int n_in,
                              void* d_out, int out_size, void* d_ws, size_t ws_size,
                              hipStream_t stream) {
    const float*     x     = (const float*)d_in[0];   // N x 512
    const float*     sf    = (const float*)d_in[1];   // N x 1
    const float*     W1    = (const float*)d_in[2];   // 512 x 8
    const float*     b1    = (const float*)d_in[3];   // 8
    const float*     gamma = (const float*)d_in[4];   // 8
    const float*     beta  = (const float*)d_in[5];   // 8
    const float*     W2    = (const float*)d_in[6];   // 8 x 512
    const float*     b2    = (const float*)d_in[7];   // 512
    const long long* ei    = (const long long*)d_in[8]; // 2 x E (int64)

    const int N = in_sizes[1];
    const int E = in_sizes[8] / 2;

    float* ws   = (float*)d_ws;
    float* dinv = ws;                       // N   (deg, then rsqrt in place)
    float* h1   = ws + (size_t)N;           // 8N  (x @ W1)
    float* agg1 = ws + (size_t)9 * N;       // 8N
    float* z    = ws + (size_t)17 * N;      // 8N
    float* agg2 = ws + (size_t)25 * N;      // 8N

    const int blk = 256;
    init_kernel<<<(8 * N + blk - 1) / blk, blk, 0, stream>>>(dinv, agg1, agg2, N);
    deg_kernel<<<(E + blk - 1) / blk, blk, 0, stream>>>(ei, dinv, E);
    dinv_kernel<<<(N + blk - 1) / blk, blk, 0, stream>>>(dinv, N);

    const int tiles = (N + 15) / 16;                // 16-row WMMA tiles
    const int gblocks = (tiles + 7) / 8;            // 8 waves / block
    gemm1_wmma_kernel<<<gblocks, blk, 0, stream>>>(x, W1, h1, N);

    scatter_kernel<<<(8 * E + blk - 1) / blk, blk, 0, stream>>>(ei, dinv, h1, agg1, E);
    ln_relu_kernel<<<(N + blk - 1) / blk, blk, 0, stream>>>(agg1, h1, dinv, b1,
                                                            gamma, beta, z, N);
    scatter_kernel<<<(8 * E + blk - 1) / blk, blk, 0, stream>>>(ei, dinv, z, agg2, E);

    gemm2_wmma_kernel<<<gblocks, blk, 0, stream>>>(agg2, z, dinv, W2, b2, sf,
                                                   (float*)d_out, N);
}